// InductiveLayer_15221364097568
// MI455X (gfx1250) — compile-verified
//
#include <hip/hip_runtime.h>

typedef __attribute__((ext_vector_type(2))) float v2f;
typedef __attribute__((ext_vector_type(8))) float v8f;

#define FDIM 256
#define DDIM 128
#define HOPS 3

// C[M,DDIM] = scale * (A[M,KDIM] @ B[KDIM,DDIM])
// Block: 256 threads = 8 waves; block tile = 32 rows x 128 cols.
// Wave w owns cols [16w,16w+16) and keeps TWO 16x16 f32 accumulators
// (rows [row0,row0+16) and [row0+16,row0+32)) that share each B fragment,
// halving global B traffic vs a 16-row block.
// A tile (32 x KDIM) staged in LDS with padded stride (KDIM+2) -> conflict-free
// ds_load_b64 A-fragment reads. Uses V_WMMA_F32_16X16X4_F32, K-loop of KDIM/4.
template <int KDIM>
__global__ __launch_bounds__(256) void gemm_wmma_kernel(
    const float* __restrict__ A, const float* __restrict__ B,
    float* __restrict__ C, const float* __restrict__ scale_ptr) {
  __shared__ float Atile[32 * (KDIM + 2)];
  const int row0 = blockIdx.x * 32;
  const int tid = threadIdx.x;

  // Cooperative, coalesced load of the 32 x KDIM A tile into LDS.
  for (int idx = tid; idx < 32 * KDIM; idx += 256) {
    const int r = idx / KDIM;
    const int c = idx % KDIM;
    Atile[r * (KDIM + 2) + c] = A[(size_t)(row0 + r) * KDIM + c];
  }
  __syncthreads();

  const int wave = tid >> 5;
  const int lane = tid & 31;
  const int col0 = wave * 16;
  const int half = lane >> 4;   // 0: lanes 0-15, 1: lanes 16-31
  const int mn = lane & 15;     // M for A-frag, N for B/C/D-frags
  const int koff = half * 2;    // K sub-offset within a 4-wide K step

  v8f acc0 = {};
  v8f acc1 = {};
  const float* arow0 = &Atile[mn * (KDIM + 2)];
  const float* arow1 = &Atile[(mn + 16) * (KDIM + 2)];

#pragma unroll 8
  for (int kk = 0; kk < KDIM / 4; ++kk) {
    const int k0 = kk * 4 + koff;
    // A 16x4 frags: lane holds A[m][k0], A[m][k0+1] (8B-aligned LDS pairs)
    v2f a0 = *(const v2f*)(arow0 + k0);
    v2f a1 = *(const v2f*)(arow1 + k0);
    // B 4x16 frag: lane holds B[k0][col0+mn], B[k0+1][col0+mn] (shared by both tiles)
    v2f b;
    b.x = B[(size_t)k0 * DDIM + col0 + mn];
    b.y = B[(size_t)(k0 + 1) * DDIM + col0 + mn];
    acc0 = __builtin_amdgcn_wmma_f32_16x16x4_f32(
        false, a0, false, b, (short)0, acc0, false, false);
    acc1 = __builtin_amdgcn_wmma_f32_16x16x4_f32(
        false, a1, false, b, (short)0, acc1, false, false);
  }

  const float scale = scale_ptr ? *scale_ptr : 1.0f;
#pragma unroll
  for (int v = 0; v < 8; ++v) {
    const int m = v + half * 8;  // C/D layout: VGPR v holds M=v (lanes 0-15) / M=v+8 (lanes 16-31)
    C[(size_t)(row0 + m) * DDIM + col0 + mn] = acc0[v] * scale;
    C[(size_t)(row0 + 16 + m) * DDIM + col0 + mn] = acc1[v] * scale;
  }
}

// One wave per edge: out[row][:] += val * fw[col][:] (128 channels = 32 lanes x float4).
__global__ __launch_bounds__(256) void edge_spmm_kernel(
    const int* __restrict__ rows, const int* __restrict__ cols,
    const float* __restrict__ vals, const float* __restrict__ fw,
    float* __restrict__ out, int E) {
  const int wid = blockIdx.x * 8 + (threadIdx.x >> 5);
  if (wid >= E) return;
  const int lane = threadIdx.x & 31;
  const int r = rows[wid];
  const int c = cols[wid];
  const float v = vals[wid];
  const float4 f = *((const float4*)(fw + (size_t)c * DDIM) + lane);
  float* dst = out + (size_t)r * DDIM + lane * 4;
  unsafeAtomicAdd(dst + 0, v * f.x);
  unsafeAtomicAdd(dst + 1, v * f.y);
  unsafeAtomicAdd(dst + 2, v * f.z);
  unsafeAtomicAdd(dst + 3, v * f.w);
}

// Wsum = W_embK[0] + W_embK[1] + W_embK[2]  (128x128)
__global__ void wsum_kernel(const float* __restrict__ WembK, float* __restrict__ Wsum) {
  const int i = blockIdx.x * blockDim.x + threadIdx.x;
  if (i < DDIM * DDIM) {
    Wsum[i] = WembK[i] + WembK[DDIM * DDIM + i] + WembK[2 * DDIM * DDIM + i];
  }
}

__global__ void relu_kernel(float* __restrict__ out, int n4) {
  const int i = blockIdx.x * blockDim.x + threadIdx.x;
  if (i < n4) {
    float4* p = (float4*)out + i;
    float4 v = *p;
    v.x = fmaxf(v.x, 0.0f);
    v.y = fmaxf(v.y, 0.0f);
    v.z = fmaxf(v.z, 0.0f);
    v.w = fmaxf(v.w, 0.0f);
    *p = v;
  }
}

extern "C" void kernel_launch(void* const* d_in, const int* in_sizes, int n_in,
                              void* d_out, int out_size, void* d_ws, size_t ws_size,
                              hipStream_t stream) {
  const float* X       = (const float*)d_in[0];  // [N, 256]
  const int*   erows   = (const int*)d_in[1];    // [3, E]
  const int*   ecols   = (const int*)d_in[2];    // [3, E]
  const float* evals   = (const float*)d_in[3];  // [3, E]
  const float* W_embed = (const float*)d_in[4];  // [256, 128]
  const float* W_feat  = (const float*)d_in[5];  // [3, 256, 128]
  const float* W_embK  = (const float*)d_in[6];  // [3, 128, 128]
  const float* alpha   = (const float*)d_in[7];  // scalar

  const int Nn = in_sizes[0] / FDIM;   // 100000 (multiple of 32)
  const int E  = in_sizes[1] / HOPS;   // 1600000

  float* out     = (float*)d_out;
  float* learned = (float*)d_ws;                       // N*128
  float* fw      = learned + (size_t)Nn * DDIM;        // N*128 (reused per hop)
  float* Wsum    = fw + (size_t)Nn * DDIM;             // 128*128

  const int mtiles  = (Nn + 31) / 32;  // 3125
  const int eblocks = (E + 7) / 8;

  // 1) Wsum = sum_h W_embK[h]
  wsum_kernel<<<(DDIM * DDIM + 255) / 256, 256, 0, stream>>>(W_embK, Wsum);
  // 2) learned = X @ W_embed
  gemm_wmma_kernel<FDIM><<<mtiles, 256, 0, stream>>>(X, W_embed, learned, nullptr);
  // 3) out = alpha * (learned @ Wsum)   [initializes d_out; fuses the 3 learned_h GEMMs]
  gemm_wmma_kernel<DDIM><<<mtiles, 256, 0, stream>>>(learned, Wsum, out, alpha);
  // 4) per hop: fw = X @ W_feat[h]; out += A_h @ fw (COO gather + atomic scatter-add)
  for (int h = 0; h < HOPS; ++h) {
    gemm_wmma_kernel<FDIM><<<mtiles, 256, 0, stream>>>(
        X, W_feat + (size_t)h * FDIM * DDIM, fw, nullptr);
    edge_spmm_kernel<<<eblocks, 256, 0, stream>>>(
        erows + (size_t)h * E, ecols + (size_t)h * E, evals + (size_t)h * E, fw, out, E);
  }
  // 5) out = relu(out)
  const int n4 = (Nn * DDIM) / 4;
  relu_kernel<<<(n4 + 255) / 256, 256, 0, stream>>>(out, n4);
}